// SlotAttention_40638980554961
// MI455X (gfx1250) — compile-verified
//
#include <hip/hip_runtime.h>
#include <math.h>

// Problem constants (match reference).
#define BB 32
#define TT 4096
#define DD 512
#define CHUNKS 32                      // T-chunks per batch row
#define ROWS_PER_CHUNK (TT / CHUNKS)   // 128
#define NWAVES 8                       // 256 threads / wave32
#define ROWS_PER_WAVE (ROWS_PER_CHUNK / NWAVES) // 16

// ---- gfx1250 async global->LDS copy (ASYNCcnt-tracked) --------------------
#if defined(__has_builtin)
#  if __has_builtin(__builtin_amdgcn_global_load_async_to_lds_b128)
#    define HAVE_ASYNC_B128 1
#  endif
#  if __has_builtin(__builtin_amdgcn_s_wait_asynccnt)
#    define HAVE_WAIT_ASYNC 1
#  endif
#endif
#ifndef HAVE_ASYNC_B128
#  define HAVE_ASYNC_B128 0
#endif
#ifndef HAVE_WAIT_ASYNC
#  define HAVE_WAIT_ASYNC 0
#endif

typedef int vint4 __attribute__((vector_size(16)));
typedef __attribute__((address_space(1))) vint4* gbl_vint4_ptr;
typedef __attribute__((address_space(3))) vint4* lds_vint4_ptr;

__device__ __forceinline__ void wait_asynccnt_zero() {
#if HAVE_WAIT_ASYNC
  __builtin_amdgcn_s_wait_asynccnt(0);
#else
  asm volatile("s_wait_asynccnt 0" ::: "memory");
#endif
}

// ---------------------------------------------------------------------------
// Kernel 1: fused scores + online softmax + weighted accumulation, one pass
// over inp. Grid = B*CHUNKS blocks of 256 threads (8 wave32s). Each wave
// handles 16 consecutive rows, two at a time (8 b128 loads in flight, one
// softmax rescale per pair). Each lane owns a 16-float D-slice.
// Workspace layout (floats):
//   wm[B*CHUNKS]   running max per (b,chunk)
//   wsum[B*CHUNKS] running sum-of-exp (scaled to wm)
//   wc[B*CHUNKS][D] unnormalized context partials (scaled to wm)
// ---------------------------------------------------------------------------
__global__ void __launch_bounds__(256)
slot_attn_partial(const float* __restrict__ inp, const int* __restrict__ lens,
                  const float* __restrict__ emb, float* __restrict__ ws) {
  __shared__ float lds_emb[DD];            // 2 KB, shared by all waves
  __shared__ float lds_c[NWAVES][DD];      // 16 KB wave-partial contexts
  __shared__ float lds_m[NWAVES];
  __shared__ float lds_s[NWAVES];

  const int tid   = threadIdx.x;
  const int lane  = tid & 31;
  const int wave  = tid >> 5;
  const int bc    = blockIdx.x;
  const int b     = bc / CHUNKS;
  const int chunk = bc % CHUNKS;
  const int t0    = chunk * ROWS_PER_CHUNK + wave * ROWS_PER_WAVE;
  const int len   = lens[b];

  // Stage the slot embedding into LDS with CDNA5 async copies (128 lanes x 16B).
  if (tid < DD / 4) {
#if HAVE_ASYNC_B128
    __builtin_amdgcn_global_load_async_to_lds_b128(
        (gbl_vint4_ptr)(emb + tid * 4),
        (lds_vint4_ptr)(&lds_emb[tid * 4]),
        0, 0);
#else
    *(float4*)(&lds_emb[tid * 4]) = *(const float4*)(emb + tid * 4);
#endif
  }
#if HAVE_ASYNC_B128
  wait_asynccnt_zero();   // each wave drains its own async loads pre-barrier
#endif
  __syncthreads();

  // Lane's 16-float embedding slice: d = j*128 + lane*4 .. +3
  const float4 e0 = *(const float4*)(&lds_emb[0 * 128 + lane * 4]);
  const float4 e1 = *(const float4*)(&lds_emb[1 * 128 + lane * 4]);
  const float4 e2 = *(const float4*)(&lds_emb[2 * 128 + lane * 4]);
  const float4 e3 = *(const float4*)(&lds_emb[3 * 128 + lane * 4]);

  float m = -INFINITY, s = 0.0f;
  float4 c0 = {0.f, 0.f, 0.f, 0.f};
  float4 c1 = {0.f, 0.f, 0.f, 0.f};
  float4 c2 = {0.f, 0.f, 0.f, 0.f};
  float4 c3 = {0.f, 0.f, 0.f, 0.f};

  // Number of valid rows for this wave's strip.
  int nv = len - t0;
  if (nv > ROWS_PER_WAVE) nv = ROWS_PER_WAVE;
  if (nv < 0) nv = 0;

  const float* rowbase = inp + ((size_t)b * TT + (size_t)t0) * DD;
  int r = 0;
  // Main loop: two rows per iteration (8 independent b128 loads in flight).
  for (; r + 2 <= nv; r += 2) {
    const float* rp0 = rowbase + (size_t)r * DD;
    const float* rp1 = rp0 + DD;
    const float4 x0 = *(const float4*)(rp0 + 0 * 128 + lane * 4);
    const float4 x1 = *(const float4*)(rp0 + 1 * 128 + lane * 4);
    const float4 x2 = *(const float4*)(rp0 + 2 * 128 + lane * 4);
    const float4 x3 = *(const float4*)(rp0 + 3 * 128 + lane * 4);
    const float4 y0 = *(const float4*)(rp1 + 0 * 128 + lane * 4);
    const float4 y1 = *(const float4*)(rp1 + 1 * 128 + lane * 4);
    const float4 y2 = *(const float4*)(rp1 + 2 * 128 + lane * 4);
    const float4 y3 = *(const float4*)(rp1 + 3 * 128 + lane * 4);

    float p0 = x0.x * e0.x + x0.y * e0.y + x0.z * e0.z + x0.w * e0.w
             + x1.x * e1.x + x1.y * e1.y + x1.z * e1.z + x1.w * e1.w
             + x2.x * e2.x + x2.y * e2.y + x2.z * e2.z + x2.w * e2.w
             + x3.x * e3.x + x3.y * e3.y + x3.z * e3.z + x3.w * e3.w;
    float p1 = y0.x * e0.x + y0.y * e0.y + y0.z * e0.z + y0.w * e0.w
             + y1.x * e1.x + y1.y * e1.y + y1.z * e1.z + y1.w * e1.w
             + y2.x * e2.x + y2.y * e2.y + y2.z * e2.z + y2.w * e2.w
             + y3.x * e3.x + y3.y * e3.y + y3.z * e3.z + y3.w * e3.w;
#pragma unroll
    for (int off = 16; off; off >>= 1) {
      p0 += __shfl_xor(p0, off, 32);   // two independent chains -> 2x ILP
      p1 += __shfl_xor(p1, off, 32);
    }

    // Online softmax update for the pair (single rescale).
    const float mn    = fmaxf(m, fmaxf(p0, p1));
    const float alpha = __expf(m - mn);  // exp(-inf)=0 on first valid pair
    const float w0    = __expf(p0 - mn);
    const float w1    = __expf(p1 - mn);
    s = s * alpha + w0 + w1;
    c0.x = c0.x * alpha + w0 * x0.x + w1 * y0.x;
    c0.y = c0.y * alpha + w0 * x0.y + w1 * y0.y;
    c0.z = c0.z * alpha + w0 * x0.z + w1 * y0.z;
    c0.w = c0.w * alpha + w0 * x0.w + w1 * y0.w;
    c1.x = c1.x * alpha + w0 * x1.x + w1 * y1.x;
    c1.y = c1.y * alpha + w0 * x1.y + w1 * y1.y;
    c1.z = c1.z * alpha + w0 * x1.z + w1 * y1.z;
    c1.w = c1.w * alpha + w0 * x1.w + w1 * y1.w;
    c2.x = c2.x * alpha + w0 * x2.x + w1 * y2.x;
    c2.y = c2.y * alpha + w0 * x2.y + w1 * y2.y;
    c2.z = c2.z * alpha + w0 * x2.z + w1 * y2.z;
    c2.w = c2.w * alpha + w0 * x2.w + w1 * y2.w;
    c3.x = c3.x * alpha + w0 * x3.x + w1 * y3.x;
    c3.y = c3.y * alpha + w0 * x3.y + w1 * y3.y;
    c3.z = c3.z * alpha + w0 * x3.z + w1 * y3.z;
    c3.w = c3.w * alpha + w0 * x3.w + w1 * y3.w;
    m = mn;
  }
  // Tail: at most one remaining row.
  if (r < nv) {
    const float* rp = rowbase + (size_t)r * DD;
    const float4 x0 = *(const float4*)(rp + 0 * 128 + lane * 4);
    const float4 x1 = *(const float4*)(rp + 1 * 128 + lane * 4);
    const float4 x2 = *(const float4*)(rp + 2 * 128 + lane * 4);
    const float4 x3 = *(const float4*)(rp + 3 * 128 + lane * 4);
    float p = x0.x * e0.x + x0.y * e0.y + x0.z * e0.z + x0.w * e0.w
            + x1.x * e1.x + x1.y * e1.y + x1.z * e1.z + x1.w * e1.w
            + x2.x * e2.x + x2.y * e2.y + x2.z * e2.z + x2.w * e2.w
            + x3.x * e3.x + x3.y * e3.y + x3.z * e3.z + x3.w * e3.w;
#pragma unroll
    for (int off = 16; off; off >>= 1) p += __shfl_xor(p, off, 32);
    const float mn    = fmaxf(m, p);
    const float alpha = __expf(m - mn);
    const float w     = __expf(p - mn);
    s = s * alpha + w;
    c0.x = c0.x * alpha + w * x0.x;  c0.y = c0.y * alpha + w * x0.y;
    c0.z = c0.z * alpha + w * x0.z;  c0.w = c0.w * alpha + w * x0.w;
    c1.x = c1.x * alpha + w * x1.x;  c1.y = c1.y * alpha + w * x1.y;
    c1.z = c1.z * alpha + w * x1.z;  c1.w = c1.w * alpha + w * x1.w;
    c2.x = c2.x * alpha + w * x2.x;  c2.y = c2.y * alpha + w * x2.y;
    c2.z = c2.z * alpha + w * x2.z;  c2.w = c2.w * alpha + w * x2.w;
    c3.x = c3.x * alpha + w * x3.x;  c3.y = c3.y * alpha + w * x3.y;
    c3.z = c3.z * alpha + w * x3.z;  c3.w = c3.w * alpha + w * x3.w;
    m = mn;
  }

  // Publish wave partials to LDS.
  *(float4*)(&lds_c[wave][0 * 128 + lane * 4]) = c0;
  *(float4*)(&lds_c[wave][1 * 128 + lane * 4]) = c1;
  *(float4*)(&lds_c[wave][2 * 128 + lane * 4]) = c2;
  *(float4*)(&lds_c[wave][3 * 128 + lane * 4]) = c3;
  if (lane == 0) { lds_m[wave] = m; lds_s[wave] = s; }
  __syncthreads();

  // Merge the 8 wave partials (redundantly per thread; trivial cost).
  float mstar = -INFINITY;
#pragma unroll
  for (int w2 = 0; w2 < NWAVES; ++w2) mstar = fmaxf(mstar, lds_m[w2]);
  float sstar = 0.0f;
  float alphas[NWAVES];
#pragma unroll
  for (int w2 = 0; w2 < NWAVES; ++w2) {
    const float mw = lds_m[w2];
    const float a  = (mw > -INFINITY) ? __expf(mw - mstar) : 0.0f;
    alphas[w2] = a;
    sstar += lds_s[w2] * a;
  }
  const int d0 = tid * 2;                 // 256 threads x 2 = 512
  float acc0 = 0.0f, acc1 = 0.0f;
#pragma unroll
  for (int w2 = 0; w2 < NWAVES; ++w2) {
    acc0 = fmaf(lds_c[w2][d0 + 0], alphas[w2], acc0);
    acc1 = fmaf(lds_c[w2][d0 + 1], alphas[w2], acc1);
  }

  float* wm   = ws;
  float* wsum = ws + (BB * CHUNKS);
  float* wc   = ws + 2 * (BB * CHUNKS);
  wc[(size_t)bc * DD + d0 + 0] = acc0;
  wc[(size_t)bc * DD + d0 + 1] = acc1;
  if (tid == 0) { wm[bc] = mstar; wsum[bc] = sstar; }
}

// ---------------------------------------------------------------------------
// Kernel 2: merge the 32 chunk partials per batch row, normalize, write out.
// Grid = B blocks of 256 threads. Reads ~2 MB total: negligible.
// ---------------------------------------------------------------------------
__global__ void __launch_bounds__(256)
slot_attn_reduce(const float* __restrict__ ws, float* __restrict__ out) {
  const int b   = blockIdx.x;
  const int tid = threadIdx.x;
  const float* wm   = ws;
  const float* wsum = ws + (BB * CHUNKS);
  const float* wc   = ws + 2 * (BB * CHUNKS);
  const float* pm = wm + b * CHUNKS;
  const float* psv = wsum + b * CHUNKS;

  float mstar = -INFINITY;
#pragma unroll
  for (int i = 0; i < CHUNKS; ++i) mstar = fmaxf(mstar, pm[i]);
  float sstar = 0.0f;
#pragma unroll
  for (int i = 0; i < CHUNKS; ++i) {
    const float mi = pm[i];
    if (mi > -INFINITY) sstar += psv[i] * __expf(mi - mstar);
  }

  const int d0 = tid * 2;
  float a0 = 0.0f, a1 = 0.0f;
  for (int i = 0; i < CHUNKS; ++i) {
    const float mi = pm[i];
    if (mi > -INFINITY) {
      const float al = __expf(mi - mstar);
      const float* cp = wc + ((size_t)(b * CHUNKS + i)) * DD + d0;
      a0 = fmaf(cp[0], al, a0);
      a1 = fmaf(cp[1], al, a1);
    }
  }
  const float inv = 1.0f / sstar;        // every batch row has >=1 valid t
  out[(size_t)b * DD + d0 + 0] = a0 * inv;
  out[(size_t)b * DD + d0 + 1] = a1 * inv;
}

extern "C" void kernel_launch(void* const* d_in, const int* in_sizes, int n_in,
                              void* d_out, int out_size, void* d_ws, size_t ws_size,
                              hipStream_t stream) {
  (void)in_sizes; (void)n_in; (void)out_size; (void)ws_size;
  const float* inp  = (const float*)d_in[0];
  const int*   lens = (const int*)d_in[1];
  const float* emb  = (const float*)d_in[2];
  float* out = (float*)d_out;
  float* ws  = (float*)d_ws;   // needs (2*1024 + 1024*512)*4 ~= 2.01 MB

  slot_attn_partial<<<dim3(BB * CHUNKS), dim3(256), 0, stream>>>(inp, lens, emb, ws);
  slot_attn_reduce<<<dim3(BB), dim3(256), 0, stream>>>(ws, out);
}